// LocalAttentionMechanism_13486197309587
// MI455X (gfx1250) — compile-verified
//
#include <hip/hip_runtime.h>

typedef __attribute__((ext_vector_type(2))) float v2f;
typedef __attribute__((ext_vector_type(8))) float v8f;

constexpr int SEQ = 2048;
constexpr int DIM = 512;
constexpr int NH  = 8;
constexpr int DKH = 64;          // head dim
constexpr int WIN = 64;          // window radius (keys i-64..i)
constexpr int SD  = SEQ * DIM;   // elements per (S,D) buffer

#define WMMA_F32(a, b, c) \
  __builtin_amdgcn_wmma_f32_16x16x4_f32(false, (a), false, (b), (short)0, (c), false, false)

// ---------------------------------------------------------------------------
// GEMM Out = X @ Wm^T with 32x64 register blocking: one wave computes
// 2 m-tiles x 4 n-tiles (8 accumulators). Per k-step: 2 A frags + 4 B frags
// feed 8 wmma. f32 WMMA 16x16x4 fragment layout:
//   A frag lane(l,half): a = X[m0+l][kk+2*half (+1)]
//   B frag lane(l,half): b = Wm[n0+l][kk+2*half (+1)]   (B[k][n] = Wm[n][k])
//   C/D: vgpr v, lane(l,half) -> D[m0 + v + 8*half][n0 + l]
// ---------------------------------------------------------------------------
__global__ __launch_bounds__(256) void qkv_proj(
    const float* __restrict__ q_in, const float* __restrict__ k_in,
    const float* __restrict__ v_in,
    const float* __restrict__ Wq, const float* __restrict__ Wk,
    const float* __restrict__ Wv,
    float* __restrict__ Qo, float* __restrict__ Ko, float* __restrict__ Vo) {
  const int wave = blockIdx.x * (blockDim.x >> 5) + (threadIdx.x >> 5);
  const int lane = threadIdx.x & 31;
  const int half = lane >> 4;
  const int l    = lane & 15;

  const int NSTR = (DIM / 16) / 4;   // 8  n-strips of 64 cols
  const int MSTR = (SEQ / 16) / 2;   // 64 m-strips of 32 rows
  int t = wave;
  const int mat = t / (MSTR * NSTR);
  t -= mat * MSTR * NSTR;
  const int ms = t / NSTR;
  const int ns = t - ms * NSTR;

  const float* X  = (mat == 0) ? q_in : (mat == 1) ? k_in : v_in;
  const float* Wm = (mat == 0) ? Wq   : (mat == 1) ? Wk   : Wv;
  float*       O  = (mat == 0) ? Qo   : (mat == 1) ? Ko   : Vo;

  const int m0 = ms * 32, n0 = ns * 64;
  const float* arow[2];
  const float* brow[4];
#pragma unroll
  for (int mi = 0; mi < 2; ++mi)
    arow[mi] = X + (size_t)(m0 + 16 * mi + l) * DIM + 2 * half;
#pragma unroll
  for (int ni = 0; ni < 4; ++ni)
    brow[ni] = Wm + (size_t)(n0 + 16 * ni + l) * DIM + 2 * half;

  v8f acc[2][4] = {};
  for (int kk = 0; kk < DIM; kk += 4) {
    v2f a[2], b[4];
#pragma unroll
    for (int mi = 0; mi < 2; ++mi) a[mi] = (v2f){ arow[mi][kk], arow[mi][kk + 1] };
#pragma unroll
    for (int ni = 0; ni < 4; ++ni) b[ni] = (v2f){ brow[ni][kk], brow[ni][kk + 1] };
#pragma unroll
    for (int mi = 0; mi < 2; ++mi)
#pragma unroll
      for (int ni = 0; ni < 4; ++ni)
        acc[mi][ni] = WMMA_F32(a[mi], b[ni], acc[mi][ni]);
  }
#pragma unroll
  for (int mi = 0; mi < 2; ++mi)
#pragma unroll
    for (int ni = 0; ni < 4; ++ni)
#pragma unroll
      for (int v = 0; v < 8; ++v)
        O[(size_t)(m0 + 16 * mi + v + 8 * half) * DIM + n0 + 16 * ni + l] = acc[mi][ni][v];
}

// ---------------------------------------------------------------------------
// Sliding-window attention. One wave per (head, 16-query tile).
// Keys span [i0-64, i0+15] -> 5 chunks of 16. Scores (Q frag reused across all
// 5 K chunks) -> LDS -> softmax (+decay bias, causal mask) -> attn_w output ->
// PV via WMMA (probs frag reused across 4 V tiles) -> att (S x D).
// ---------------------------------------------------------------------------
__global__ __launch_bounds__(256) void local_attn(
    const float* __restrict__ Q, const float* __restrict__ K,
    const float* __restrict__ V,
    float* __restrict__ att, float* __restrict__ attn_w) {
  __shared__ float smem[8][16][81];   // 8 waves/block, 16 rows, 80 keys (+pad)

  const int wslot = threadIdx.x >> 5;
  const int wave  = blockIdx.x * (blockDim.x >> 5) + wslot;
  const int lane  = threadIdx.x & 31;
  const int half  = lane >> 4;
  const int l     = lane & 15;

  const int QT = SEQ / 16;            // 128 query tiles
  const int h  = wave / QT;
  const int qt = wave - h * QT;
  const int i0 = qt * 16;
  const int kbase0 = i0 - WIN;        // key index at p = 0

  const float* Qh = Q + h * DKH;
  const float* Kh = K + h * DKH;
  const float* Vh = V + h * DKH;

  // -------- scores: 5 chunks, Q frag loaded once per k-step --------
  const float* arow = Qh + (size_t)(i0 + l) * DIM + 2 * half;
  const float* krow[5];
#pragma unroll
  for (int c = 0; c < 5; ++c) {
    int kr = kbase0 + c * 16 + l;
    if (kr < 0) kr = 0;               // clamped; masked in softmax
    krow[c] = Kh + (size_t)kr * DIM + 2 * half;
  }
  v8f sc[5] = {};
  for (int kk = 0; kk < DKH; kk += 4) {
    v2f a = { arow[kk], arow[kk + 1] };
#pragma unroll
    for (int c = 0; c < 5; ++c) {
      v2f b = { krow[c][kk], krow[c][kk + 1] };
      sc[c] = WMMA_F32(a, b, sc[c]);
    }
  }
#pragma unroll
  for (int c = 0; c < 5; ++c)
#pragma unroll
    for (int v = 0; v < 8; ++v)
      smem[wslot][v + 8 * half][c * 16 + l] = sc[c][v];
  __syncthreads();

  // -------- softmax per query row (lanes 0..15, one row each) --------
  if (lane < 16) {
    const int q = lane;
    const int i = i0 + q;
    float* row = smem[wslot][q];
    float mx = -3.0e38f;
    for (int r = 0; r <= WIN; ++r) {
      if (i - WIN + r >= 0) {
        float s = row[q + r] * 0.125f + 0.1f * __expf(-0.1f * (float)(WIN - r));
        mx = fmaxf(mx, s);
      }
    }
    float sum = 0.f;
    for (int p = 0; p < 80; ++p) {
      const int r = p - q;
      float e = 0.f;
      if (r >= 0 && r <= WIN && (i - WIN + r) >= 0) {
        e = __expf(row[p] * 0.125f + 0.1f * __expf(-0.1f * (float)(WIN - r)) - mx);
      }
      row[p] = e;
      sum += e;
    }
    const float inv = 1.f / sum;
    for (int p = 0; p < 80; ++p) row[p] *= inv;
  }
  __syncthreads();

  // -------- attn_w output: 16 rows x 128 slots --------
  for (int idx = lane; idx < 16 * 128; idx += 32) {
    const int qrow = idx >> 7;
    const int slot = idx & 127;
    const int i    = i0 + qrow;
    const int win  = min(SEQ, i + WIN + 1) - max(0, i - WIN);
    const bool stored = (win <= 2 * WIN);
    const int ridx = slot + max(WIN - i, 0);
    const bool vm  = (ridx <= WIN) && (slot < win) && stored;
    float val = 0.f;
    if (vm) val = smem[wslot][qrow][qrow + ridx];   // probs_ref[ridx]
    attn_w[((size_t)h * SEQ + i) * 128 + slot] = val;
  }
  __syncthreads();

  // -------- out tile: O(16x64) = sum_c P_c(16x16) x V_c(16x64) --------
  v8f o[4] = {};
#pragma unroll
  for (int c = 0; c < 5; ++c) {
    const int kb = kbase0 + c * 16;
#pragma unroll
    for (int kk = 0; kk < 16; kk += 4) {
      const int k0 = kb + kk + 2 * half;
      const int kr0 = (k0 < 0) ? 0 : k0;            // probs==0 for invalid keys
      const int kr1 = (k0 + 1 < 0) ? 0 : k0 + 1;
      v2f a = { smem[wslot][l][c * 16 + kk + 2 * half],
                smem[wslot][l][c * 16 + kk + 2 * half + 1] };
      const float* vr0 = Vh + (size_t)kr0 * DIM;
      const float* vr1 = Vh + (size_t)kr1 * DIM;
#pragma unroll
      for (int ni = 0; ni < 4; ++ni) {
        v2f b = { vr0[16 * ni + l], vr1[16 * ni + l] };
        o[ni] = WMMA_F32(a, b, o[ni]);
      }
    }
  }
#pragma unroll
  for (int v = 0; v < 8; ++v) {
    float* orow = att + (size_t)(i0 + v + 8 * half) * DIM + h * DKH + l;
#pragma unroll
    for (int ni = 0; ni < 4; ++ni) orow[16 * ni] = o[ni][v];
  }
}

// ---------------------------------------------------------------------------
// output = att @ Wo^T + bo, 32x64 register blocking like qkv_proj
// ---------------------------------------------------------------------------
__global__ __launch_bounds__(256) void out_proj(
    const float* __restrict__ att, const float* __restrict__ Wo,
    const float* __restrict__ bo, float* __restrict__ out) {
  const int wave = blockIdx.x * (blockDim.x >> 5) + (threadIdx.x >> 5);
  const int lane = threadIdx.x & 31;
  const int half = lane >> 4;
  const int l    = lane & 15;

  const int NSTR = (DIM / 16) / 4;   // 8
  const int ms = wave / NSTR;
  const int ns = wave - ms * NSTR;
  const int m0 = ms * 32, n0 = ns * 64;

  const float* arow[2];
  const float* brow[4];
#pragma unroll
  for (int mi = 0; mi < 2; ++mi)
    arow[mi] = att + (size_t)(m0 + 16 * mi + l) * DIM + 2 * half;
#pragma unroll
  for (int ni = 0; ni < 4; ++ni)
    brow[ni] = Wo + (size_t)(n0 + 16 * ni + l) * DIM + 2 * half;

  v8f acc[2][4] = {};
  for (int kk = 0; kk < DIM; kk += 4) {
    v2f a[2], b[4];
#pragma unroll
    for (int mi = 0; mi < 2; ++mi) a[mi] = (v2f){ arow[mi][kk], arow[mi][kk + 1] };
#pragma unroll
    for (int ni = 0; ni < 4; ++ni) b[ni] = (v2f){ brow[ni][kk], brow[ni][kk + 1] };
#pragma unroll
    for (int mi = 0; mi < 2; ++mi)
#pragma unroll
      for (int ni = 0; ni < 4; ++ni)
        acc[mi][ni] = WMMA_F32(a[mi], b[ni], acc[mi][ni]);
  }
#pragma unroll
  for (int ni = 0; ni < 4; ++ni) {
    const float bias = bo[n0 + 16 * ni + l];
#pragma unroll
    for (int mi = 0; mi < 2; ++mi)
#pragma unroll
      for (int v = 0; v < 8; ++v)
        out[(size_t)(m0 + 16 * mi + v + 8 * half) * DIM + n0 + 16 * ni + l] =
            acc[mi][ni][v] + bias;
  }
}

// ---------------------------------------------------------------------------
extern "C" void kernel_launch(void* const* d_in, const int* in_sizes, int n_in,
                              void* d_out, int out_size, void* d_ws, size_t ws_size,
                              hipStream_t stream) {
  const float* q_in = (const float*)d_in[0];
  const float* k_in = (const float*)d_in[1];
  const float* v_in = (const float*)d_in[2];
  const float* Wq   = (const float*)d_in[3];
  const float* Wk   = (const float*)d_in[4];
  const float* Wv   = (const float*)d_in[5];
  const float* Wo   = (const float*)d_in[6];
  const float* bo   = (const float*)d_in[7];

  float* out    = (float*)d_out;            // (1, S, D)
  float* attn_w = out + (size_t)SD;         // (1, H, S, 2W)

  float* Qb   = (float*)d_ws;               // 4 x 4MB fp32 scratch
  float* Kb   = Qb + SD;
  float* Vb   = Kb + SD;
  float* attb = Vb + SD;

  // 3 matrices * 64 m-strips * 8 n-strips = 1536 waves / 8 per block
  qkv_proj<<<192, 256, 0, stream>>>(q_in, k_in, v_in, Wq, Wk, Wv, Qb, Kb, Vb);
  // 8 heads * 128 query tiles = 1024 waves / 8 per block
  local_attn<<<128, 256, 0, stream>>>(Qb, Kb, Vb, attb, attn_w);
  // 64 m-strips * 8 n-strips = 512 waves / 8 per block
  out_proj<<<64, 256, 0, stream>>>(attb, Wo, bo, out);
}